// LengthRegulator_41111426957351
// MI455X (gfx1250) — compile-verified
//
#include <hip/hip_runtime.h>
#include <hip/hip_bf16.h>
#include <cstdint>
#include <cstddef>

// Fixed by the reference setup_inputs(): B=16, T=512. D and target_len are
// derived from in_sizes/out_size at launch time.
#define LR_B   16
#define LR_T   512
#define LR_BT  256   // output rows handled per block in the gather kernel

#if defined(__gfx1250__) && \
    __has_builtin(__builtin_amdgcn_global_load_async_to_lds_b32) && \
    __has_builtin(__builtin_amdgcn_s_wait_asynccnt)
#define LR_HAVE_ASYNC 1
#else
#define LR_HAVE_ASYNC 0
#endif

typedef __attribute__((address_space(1))) int  lr_gint;
typedef __attribute__((address_space(3))) int  lr_lint;

// ---------------------------------------------------------------------------
// Kernel 1: per-batch cumulative sum of max(durations, 1).
// One wave32 per batch: each lane owns 16 contiguous durations, computes a
// local inclusive prefix, then a wave-level shuffle scan combines lane totals.
// ---------------------------------------------------------------------------
__global__ __launch_bounds__(32) void lr_scan_kernel(const int* __restrict__ dur,
                                                     int* __restrict__ cum) {
  const int b    = blockIdx.x;
  const int lane = threadIdx.x;              // 0..31 (wave32)
  const int per  = LR_T / 32;                // 16 elements per lane
  const int base = b * LR_T + lane * per;

  int vals[per];
  int s = 0;
#pragma unroll
  for (int i = 0; i < per; ++i) {
    int d = dur[base + i];
    d = d < 1 ? 1 : d;
    s += d;
    vals[i] = s;                             // local inclusive prefix
  }
  const int local_total = s;

  // Inclusive wave32 scan of per-lane totals.
#pragma unroll
  for (int off = 1; off < 32; off <<= 1) {
    int t = __shfl_up(s, off, 32);
    if (lane >= off) s += t;
  }
  const int excl = s - local_total;          // exclusive offset for this lane

#pragma unroll
  for (int i = 0; i < per; ++i) cum[base + i] = excl + vals[i];
}

// ---------------------------------------------------------------------------
// Kernel 2: gather/expand. Each block owns LR_BT consecutive output rows of
// one batch. cum[b][:] is staged in LDS (async global->LDS on gfx1250), each
// thread binary-searches its own t, then 8 waves stream the rows out with
// coalesced float4 accesses (3 float4 per lane per row for D=384).
// ---------------------------------------------------------------------------
__global__ __launch_bounds__(256) void lr_gather_kernel(
    const float* __restrict__ x, const int* __restrict__ cum,
    float* __restrict__ out, int D, int target_len) {
  __shared__ int s_cum[LR_T];
  __shared__ int s_idx[LR_BT];

  const int blocks_per_batch = (target_len + LR_BT - 1) / LR_BT;
  const int b   = blockIdx.x / blocks_per_batch;
  const int t0  = (blockIdx.x % blocks_per_batch) * LR_BT;
  const int tid = threadIdx.x;
  const int* crow = cum + b * LR_T;

  // ---- Stage this batch's cumsum row into LDS --------------------------------
#if LR_HAVE_ASYNC
  for (int i = tid; i < LR_T; i += 256) {
    __builtin_amdgcn_global_load_async_to_lds_b32(
        (lr_gint*)(crow + i),
        (lr_lint*)(&s_cum[i]),
        0, 0);
  }
  __builtin_amdgcn_s_wait_asynccnt(0);
#else
  for (int i = tid; i < LR_T; i += 256) s_cum[i] = crow[i];
#endif
  __syncthreads();

  // ---- Per-thread binary search: idx = first i with cum[i] > t ---------------
  const int total = s_cum[LR_T - 1];
  const int t     = t0 + tid;
  int lo = 0, hi = LR_T;
  while (lo < hi) {
    const int mid = (lo + hi) >> 1;
    if (s_cum[mid] <= t) lo = mid + 1; else hi = mid;
  }
  int idx = lo < (LR_T - 1) ? lo : (LR_T - 1);   // min(idx, T-1)
  s_idx[tid] = (t < total) ? idx : -1;           // -1 => write zeros
  __syncthreads();

  // ---- Row copy: one wave per row, 8 waves round-robin over LR_BT rows -------
  const int wave = tid >> 5;
  const int lane = tid & 31;
  const int nvec = D >> 2;                       // float4 per row (96 for D=384)

  for (int r = wave; r < LR_BT; r += 8) {
    const int t_r = t0 + r;
    if (t_r >= target_len) continue;
    const int id = s_idx[r];
    float4* __restrict__ dst = (float4*)(out + ((size_t)b * target_len + t_r) * (size_t)D);

    if (id >= 0) {
      const float4* __restrict__ src =
          (const float4*)(x + ((size_t)b * LR_T + id) * (size_t)D);
      // Prefetch the source row this wave will copy next (12 cachelines/row).
      if (r + 8 < LR_BT) {
        const int nid = s_idx[r + 8];
        if (nid >= 0 && lane < 12) {
          const char* nsrc = (const char*)(x + ((size_t)b * LR_T + nid) * (size_t)D);
          __builtin_prefetch(nsrc + lane * 128, 0, 0);   // global_prefetch_b8
        }
      }
#pragma unroll
      for (int j = lane; j < nvec; j += 32) dst[j] = src[j];
    } else {
      const float4 z = make_float4(0.f, 0.f, 0.f, 0.f);
#pragma unroll
      for (int j = lane; j < nvec; j += 32) dst[j] = z;
    }
  }
}

// ---------------------------------------------------------------------------
// Host launcher.
// ---------------------------------------------------------------------------
extern "C" void kernel_launch(void* const* d_in, const int* in_sizes, int n_in,
                              void* d_out, int out_size, void* d_ws, size_t ws_size,
                              hipStream_t stream) {
  const float* x   = (const float*)d_in[0];   // (B, T, D) f32
  const int*   dur = (const int*)d_in[1];     // (B, T) i32
  float*       out = (float*)d_out;           // (B, target_len, D) f32

  const int B = LR_B;
  const int T = LR_T;
  const int D = in_sizes[0] / in_sizes[1];          // 384
  const int target_len = out_size / (B * D);        // 4096

  int* cum = (int*)d_ws;                            // B*T ints (32 KB) scratch

  lr_scan_kernel<<<B, 32, 0, stream>>>(dur, cum);

  const int blocks_per_batch = (target_len + LR_BT - 1) / LR_BT;
  lr_gather_kernel<<<B * blocks_per_batch, 256, 0, stream>>>(x, cum, out, D, target_len);
  (void)n_in; (void)ws_size; (void)T;
}